// LiftSplatShoot_30348238914141
// MI455X (gfx1250) — compile-verified
//
#include <hip/hip_runtime.h>

typedef __attribute__((ext_vector_type(2))) float v2f;
typedef __attribute__((ext_vector_type(8))) float v8f;

#define NCAM 48            // B*N = 8*6
#define DPTS 41
#define FHH 8
#define FWW 22
#define PPC (DPTS*FHH*FWW)       // 7216 points per camera
#define TILES_PER_CAM (PPC/16)   // 451 (exact)
#define NTILES (NCAM*TILES_PER_CAM) // 21648
#define CCH 64
#define NX0 200
#define NX1 200
#define NBINS (8*NX0*NX1)        // 320000
#define ACC_FLOATS (NBINS*CCH)   // 20,480,000 floats = 81.92 MB (fits in L2)

// ---------------------------------------------------------------- setup ----
__device__ inline void inv3_eps(const float* __restrict__ M, float eps, float* R) {
    float a = M[0]+eps, b = M[1],     c = M[2];
    float d = M[3],     e = M[4]+eps, f = M[5];
    float g = M[6],     h = M[7],     i = M[8]+eps;
    float A =  (e*i - f*h);
    float B = -(d*i - f*g);
    float C =  (d*h - e*g);
    float inv = 1.0f / (a*A + b*B + c*C);
    R[0] = A*inv;            R[1] = -(b*i - c*h)*inv; R[2] =  (b*f - c*e)*inv;
    R[3] = B*inv;            R[4] =  (a*i - c*g)*inv; R[5] = -(a*f - c*d)*inv;
    R[6] = C*inv;            R[7] = -(a*h - b*g)*inv; R[8] =  (a*e - b*d)*inv;
}

// Per camera: store P = [inv_post | -inv_post@post_trans] (3x4 row-major)
// then Q = [rots@inv(intrins) | trans] (3x4 row-major)  -> 24 floats/cam
__global__ void lss_setup(const float* __restrict__ rots,
                          const float* __restrict__ trans,
                          const float* __restrict__ intrins,
                          const float* __restrict__ post_rots,
                          const float* __restrict__ post_trans,
                          float* __restrict__ cam_tab) {
    int k = blockIdx.x * blockDim.x + threadIdx.x;
    if (k >= NCAM) return;
    const float eps = 1e-6f;
    float ip[9]; inv3_eps(post_rots + k*9, eps, ip);
    float ii[9]; inv3_eps(intrins   + k*9, eps, ii);
    const float* R = rots + k*9;
    float Q[9];
    #pragma unroll
    for (int r = 0; r < 3; ++r)
        #pragma unroll
        for (int c = 0; c < 3; ++c)
            Q[r*3+c] = R[r*3+0]*ii[0*3+c] + R[r*3+1]*ii[1*3+c] + R[r*3+2]*ii[2*3+c];
    const float* pt = post_trans + k*3;
    const float* t  = trans      + k*3;
    float* o = cam_tab + k*24;
    #pragma unroll
    for (int r = 0; r < 3; ++r) {
        o[r*4+0] = ip[r*3+0]; o[r*4+1] = ip[r*3+1]; o[r*4+2] = ip[r*3+2];
        o[r*4+3] = -(ip[r*3+0]*pt[0] + ip[r*3+1]*pt[1] + ip[r*3+2]*pt[2]);
        o[12+r*4+0] = Q[r*3+0]; o[12+r*4+1] = Q[r*3+1]; o[12+r*4+2] = Q[r*3+2];
        o[12+r*4+3] = t[r];
    }
}

// ----------------------------------------------------------------- zero ----
__global__ void lss_zero(float4* __restrict__ acc, int n4) {
    int stride = gridDim.x * blockDim.x;
    float4 z = make_float4(0.f, 0.f, 0.f, 0.f);
    for (int i = blockIdx.x * blockDim.x + threadIdx.x; i < n4; i += stride)
        acc[i] = z;   // regular (RT) stores: accumulator lines WANT to live in L2
}

// -------------------------------------------- fused geometry + scatter ----
// One wave32 handles one 16-point tile:
//   WMMA#1: q = [inv_post | -inv_post@pt] * (u,v,d,1)
//   safe_d nonlinearity + lane repack (__shfl -> ds_bpermute)
//   WMMA#2: g = [combine | trans] * (q0*sd, q1*sd, sd, 1)
// then 64-channel atomic scatter into (bin, C)-contiguous accumulator.
// x is streamed once -> non-temporal loads keep L2 free for the accumulator.
__launch_bounds__(256)
__global__ void lss_scatter(const float* __restrict__ x,
                            const float* __restrict__ cam_tab,
                            float* __restrict__ acc) {
    __shared__ float cams[NCAM*24];
    int tid = threadIdx.x;
    for (int i = tid; i < NCAM*24; i += 256) cams[i] = cam_tab[i];
    __syncthreads();

    int wave = (blockIdx.x << 3) | (tid >> 5);
    if (wave >= NTILES) return;
    int lane = tid & 31;

    int cam   = wave / TILES_PER_CAM;
    int local = wave - cam * TILES_PER_CAM;
    int p0    = local * 16;
    int bidx  = cam / 6;                      // batch index

    long xbase = ((long)cam * PPC + p0) * CCH;
    // prefetch this tile's 4KB of features (32 lanes x 128B) -> global_prefetch_b8
    __builtin_prefetch(x + xbase + (long)lane * 32, 0, 0);

    const float* P = cams + cam * 24;
    int  m  = lane & 15;
    bool lo = lane < 16;
    int  k0 = lo ? 0 : 2;

    // A operands (16x4 f32: lanes 0-15 rows M, reg0=K(k0), reg1=K(k0+1))
    v2f A1, A2;
    A1.x = (m < 3) ? P[m*4 + k0    ] : 0.f;
    A1.y = (m < 3) ? P[m*4 + k0 + 1] : 0.f;
    A2.x = (m < 3) ? P[12 + m*4 + k0    ] : 0.f;
    A2.y = (m < 3) ? P[12 + m*4 + k0 + 1] : 0.f;

    // Frustum point for column m of this tile (flattened (D,H,W) order)
    int p   = p0 + m;
    int w   = p % FWW;
    int hq  = p / FWW;
    int h   = hq % FHH;
    int dix = hq / FHH;
    float u  = (float)w * (351.0f / 21.0f);
    float v  = (float)h * (127.0f / 7.0f);
    float dd = 4.0f + (float)dix;

    // B operand (4x16 f32: low lanes K0/K1, high lanes K2/K3; column = lane%16)
    v2f B1;
    B1.x = lo ? u : dd;
    B1.y = lo ? v : 1.0f;

    v8f zc = {};
    v8f D1 = __builtin_amdgcn_wmma_f32_16x16x4_f32(false, A1, false, B1,
                                                   (short)0, zc, false, false);
    float q0 = D1[0], q1 = D1[1], q2 = D1[2];   // lane n holds point n (n<16)

    float tq  = q2 + 1e-6f;
    float sgn = tq > 0.f ? 1.f : (tq < 0.f ? -1.f : 0.f);
    float sd  = (__builtin_fabsf(q2) < 1e-6f) ? 1e-6f * sgn : q2;

    float sd_lo = __shfl(sd, m, 32);            // high lanes fetch point (lane-16)'s sd
    v2f B2;
    B2.x = lo ? q0 * sd : sd_lo;
    B2.y = lo ? q1 * sd : 1.0f;

    v8f D2 = __builtin_amdgcn_wmma_f32_16x16x4_f32(false, A2, false, B2,
                                                   (short)0, zc, false, false);
    float g0 = D2[0], g1 = D2[1], g2 = D2[2];

    // Voxelize (int cast truncates toward zero, matching jnp astype(int32))
    float fx = (g0 + 50.0f) * 2.0f;
    float fy = (g1 + 50.0f) * 2.0f;
    float fz = (g2 + 10.0f) * 0.05f;
    int gx = (int)fx, gy = (int)fy, gz = (int)fz;
    bool kept = (gx >= 0) & (gx < NX0) & (gy >= 0) & (gy < NX1) & (gz == 0);
    int bin  = (bidx * NX0 + gx) * NX1 + gy;
    int binK = kept ? bin : -1;

    // Scatter: lanes cover channels [lane] and [lane+32]; bins are C-contiguous
    // so each point's atomic burst hits ~2 cache lines (L2-resident accumulator).
    for (int n = 0; n < 16; ++n) {
        int bb = __shfl(binK, n, 32);
        if (bb >= 0) {
            const float* xs2 = x + xbase + (long)n * CCH;
            float v0 = __builtin_nontemporal_load(xs2 + lane);       // TH=NT stream
            float v1 = __builtin_nontemporal_load(xs2 + lane + 32);  // TH=NT stream
            __hip_atomic_fetch_add(acc + (long)bb * CCH + lane,      v0,
                                   __ATOMIC_RELAXED, __HIP_MEMORY_SCOPE_AGENT);
            __hip_atomic_fetch_add(acc + (long)bb * CCH + lane + 32, v1,
                                   __ATOMIC_RELAXED, __HIP_MEMORY_SCOPE_AGENT);
        }
    }
}

// ------------------------------------------------------------ transpose ----
// (b, x, y, c) accumulator -> (b, c, 1, x, y) output; LDS tile padded to 65
// floats/row so the strided read phase is bank-conflict free. Accumulator
// reads are last-use and output stores are write-once -> both non-temporal.
__launch_bounds__(256)
__global__ void lss_transpose(const float* __restrict__ acc, float* __restrict__ out) {
    __shared__ float tile[NX1 * 65];
    int bx   = blockIdx.x;            // b*200 + x
    int b    = bx / NX0;
    int xrow = bx - b * NX0;
    const float* src = acc + ((long)b * NX0 + xrow) * NX1 * CCH;
    for (int i = threadIdx.x; i < NX1 * CCH; i += 256) {
        int y = i >> 6, c = i & 63;
        tile[y * 65 + c] = __builtin_nontemporal_load(src + i);  // coalesced, NT
    }
    __syncthreads();
    float* dst = out + (long)b * CCH * (NX0 * NX1) + (long)xrow * NX1;
    for (int i = threadIdx.x; i < NX1 * CCH; i += 256) {
        int c = i / NX1, y = i - c * NX1;
        __builtin_nontemporal_store(tile[y * 65 + c],
                                    dst + (long)c * (NX0 * NX1) + y);  // coalesced, NT
    }
}

// --------------------------------------------------------------- launch ----
extern "C" void kernel_launch(void* const* d_in, const int* in_sizes, int n_in,
                              void* d_out, int out_size, void* d_ws, size_t ws_size,
                              hipStream_t stream) {
    const float* x          = (const float*)d_in[0];
    const float* rots       = (const float*)d_in[1];
    const float* trans      = (const float*)d_in[2];
    const float* intrins    = (const float*)d_in[3];
    const float* post_rots  = (const float*)d_in[4];
    const float* post_trans = (const float*)d_in[5];
    float* out = (float*)d_out;

    float* acc     = (float*)d_ws;           // 20,480,000 floats
    float* cam_tab = acc + ACC_FLOATS;       // + 48*24 floats

    lss_zero<<<2048, 256, 0, stream>>>((float4*)acc, ACC_FLOATS / 4);
    lss_setup<<<1, 64, 0, stream>>>(rots, trans, intrins, post_rots, post_trans, cam_tab);
    lss_scatter<<<NTILES / 8, 256, 0, stream>>>(x, cam_tab, acc);
    lss_transpose<<<8 * NX0, 256, 0, stream>>>(acc, out);
}